// GraphConvExpert_70875550319091
// MI455X (gfx1250) — compile-verified
//
#include <hip/hip_runtime.h>

#define N_NODES 100000
#define N_EDGES 640000
#define HID 128
#define PITCH 132   // 128 + 4 pad: per-lane row stride hits all-distinct LDS banks

typedef float v2f __attribute__((ext_vector_type(2)));
typedef float v8f __attribute__((ext_vector_type(8)));

// ---------------------------------------------------------------------------
// Edge aggregation: agg[dst] += x[src]  (one thread per (edge, feature))
// ---------------------------------------------------------------------------
__global__ __launch_bounds__(256) void gnn_scatter_add(
    const float* __restrict__ x,
    const long long* __restrict__ src,
    const long long* __restrict__ dst,
    float* __restrict__ agg)
{
    int tid = blockIdx.x * 256 + threadIdx.x;   // < 640000*128 = 81.92M
    int e = tid >> 7;                            // edge id
    int f = tid & 127;                           // feature id
    if (e < N_EDGES) {
        int s = (int)src[e];
        int d = (int)dst[e];
        float v = x[(long long)s * HID + f];
        atomicAdd(&agg[(long long)d * HID + f], v);
    }
}

// ---------------------------------------------------------------------------
// Fused dual GEMM: out = act( xin @ Wroot^T + agg @ Wrel^T + b )
// Block = 128 node rows, 8 waves; wave w owns rows [w*16, w*16+16), all 128 cols.
// f32 WMMA 16x16x4: A frag = float2/lane (lane<16 -> K={0,1}, lane>=16 -> K={2,3})
//                   B frag = float2/lane, B[k][o] = W[o][k] read from W row o.
// ---------------------------------------------------------------------------
__global__ __launch_bounds__(256) void gnn_dual_gemm(
    const float* __restrict__ xin,    // [N,128]
    const float* __restrict__ agg,    // [N,128]
    const float* __restrict__ Wroot,  // [128,128] row-major [o][k]
    const float* __restrict__ Wrel,   // [128,128]
    const float* __restrict__ bias,   // [128]
    float* __restrict__ out,          // [N,128]
    int relu)
{
    __shared__ float a_lds[128 * PITCH];   // 67.6 KB: block's 128xK A tile
    __shared__ float w_lds[128 * PITCH];   // 67.6 KB: weight matrix

    const int t     = threadIdx.x;
    const int lane  = t & 31;
    const int wv    = t >> 5;        // wave 0..7
    const int lmod  = lane & 15;     // M (A) / output col within tile (B)
    const int lhalf = lane >> 4;     // K half selector
    const long long row_blk = (long long)blockIdx.x * 128;

    v8f c[8] = {};                    // 8 col-tiles of 16x16 f32 accum

    #pragma unroll 1
    for (int phase = 0; phase < 2; ++phase) {
        const float* A = phase ? agg  : xin;
        const float* W = phase ? Wrel : Wroot;

        __syncthreads();              // previous-phase LDS reads done
        // Stage A tile (128x128) and W (128x128): 4096 float4 each, 16/thread
        #pragma unroll
        for (int i = 0; i < 16; ++i) {
            int idx4 = t + i * 256;           // float4 index 0..4095
            int r    = idx4 >> 5;             // row 0..127
            int ccol = (idx4 & 31) << 2;      // col 0..124 step 4
            long long gr = row_blk + r;
            if (gr >= N_NODES) gr = N_NODES - 1;   // clamp: garbage rows never stored
            float4 va = *(const float4*)(A + gr * HID + ccol);
            *(float4*)&a_lds[r * PITCH + ccol] = va;
            float4 vw = *(const float4*)(W + (long long)idx4 * 4);
            *(float4*)&w_lds[r * PITCH + ccol] = vw;
        }
        __syncthreads();

        const float* a_base = &a_lds[(wv * 16 + lmod) * PITCH + lhalf * 2];
        #pragma unroll 4
        for (int kt = 0; kt < 32; ++kt) {
            const int k0 = kt * 4;
            v2f a = *(const v2f*)(a_base + k0);
            #pragma unroll
            for (int ct = 0; ct < 8; ++ct) {
                v2f b = *(const v2f*)&w_lds[(ct * 16 + lmod) * PITCH + lhalf * 2 + k0];
                c[ct] = __builtin_amdgcn_wmma_f32_16x16x4_f32(
                            false, a, false, b, (short)0, c[ct], false, false);
            }
        }
    }

    // Epilogue: bias (+ReLU), predicated store.
    // C layout: vgpr j, lanes<16 -> row j, lanes>=16 -> row j+8; col = lmod.
    #pragma unroll
    for (int ct = 0; ct < 8; ++ct) {
        const int col = ct * 16 + lmod;
        const float bv = bias[col];
        #pragma unroll
        for (int j = 0; j < 8; ++j) {
            long long gr = row_blk + wv * 16 + lhalf * 8 + j;
            if (gr < N_NODES) {
                float v = c[ct][j] + bv;
                if (relu) v = fmaxf(v, 0.0f);
                out[gr * HID + col] = v;
            }
        }
    }
}

// ---------------------------------------------------------------------------
extern "C" void kernel_launch(void* const* d_in, const int* in_sizes, int n_in,
                              void* d_out, int out_size, void* d_ws, size_t ws_size,
                              hipStream_t stream) {
    (void)in_sizes; (void)n_in; (void)out_size; (void)ws_size;

    const float*     x      = (const float*)d_in[0];
    const long long* ei     = (const long long*)d_in[1];   // int64 [2, E]
    const long long* esrc   = ei;
    const long long* edst   = ei + N_EDGES;
    const float*     Wrel0  = (const float*)d_in[2];
    const float*     b0     = (const float*)d_in[3];
    const float*     Wroot0 = (const float*)d_in[4];
    const float*     Wrel1  = (const float*)d_in[5];
    const float*     b1     = (const float*)d_in[6];
    const float*     Wroot1 = (const float*)d_in[7];
    const float*     Wrel2  = (const float*)d_in[8];
    const float*     b2     = (const float*)d_in[9];
    const float*     Wroot2 = (const float*)d_in[10];
    float*           out    = (float*)d_out;

    const size_t fbytes = (size_t)N_NODES * HID * sizeof(float);   // 51.2 MB
    char* ws = (char*)d_ws;
    float* agg = (float*)(ws);
    float* h0  = (float*)(ws + fbytes);
    float* h1  = (float*)(ws + 2 * fbytes);

    const int sgrid = (N_EDGES * HID) / 256;       // 320000
    const int ggrid = (N_NODES + 127) / 128;       // 782

    // Layer 0
    hipMemsetAsync(agg, 0, fbytes, stream);
    gnn_scatter_add<<<sgrid, 256, 0, stream>>>(x, esrc, edst, agg);
    gnn_dual_gemm<<<ggrid, 256, 0, stream>>>(x, agg, Wroot0, Wrel0, b0, h0, 1);
    // Layer 1
    hipMemsetAsync(agg, 0, fbytes, stream);
    gnn_scatter_add<<<sgrid, 256, 0, stream>>>(h0, esrc, edst, agg);
    gnn_dual_gemm<<<ggrid, 256, 0, stream>>>(h0, agg, Wroot1, Wrel1, b1, h1, 1);
    // Layer 2 (no ReLU)
    hipMemsetAsync(agg, 0, fbytes, stream);
    gnn_scatter_add<<<sgrid, 256, 0, stream>>>(h1, esrc, edst, agg);
    gnn_dual_gemm<<<ggrid, 256, 0, stream>>>(h1, agg, Wroot2, Wrel2, b2, out, 0);
}